// MiniRocketBase_47184510714287
// MI455X (gfx1250) — compile-verified
//
#include <hip/hip_runtime.h>

typedef float v2f __attribute__((ext_vector_type(2)));
typedef float v8f __attribute__((ext_vector_type(8)));

#define B_    8
#define C_    8
#define L_    2048
#define ND_   4
#define NK_   84      // kernels
#define NF_   30      // features per dilation per kernel
#define TILE_ 256     // t-values per workgroup
#define HALO_ 32      // max |(k-4)*d| = 4*8
#define XW_   (TILE_ + 2*HALO_)   // 320
#define QT_   64      // quarter of the tile (N dimension of GEMM pass)
#define KDIM_ 72      // C_ * 9
#define ASTR_ 73      // padded A row stride
#define CSTR_ 68      // padded csum row stride
#define MROWS_ 96     // 84 padded up to 6 WMMA row-tiles

// ---------------- zero workspace accumulator ----------------
__global__ __launch_bounds__(256) void mr_zero(float* __restrict__ acc, int n) {
  int i = blockIdx.x * blockDim.x + threadIdx.x;
  if (i < n) acc[i] = 0.f;
}

// ---------------- fused conv+combine (WMMA GEMM) + soft-PPV partials ----------------
__global__ __launch_bounds__(256) void mr_main(const float* __restrict__ x,
                                               const float* __restrict__ kers,
                                               const float* __restrict__ comb,
                                               const float* __restrict__ biases,
                                               float* __restrict__ acc) {
  __shared__ float sA[MROWS_ * ASTR_];   // 28032 B : fused A = comb (x) kernel taps
  __shared__ float sX[C_ * XW_];         // 10240 B : x tile + halo, zero padded
  __shared__ float sC[NK_ * CSTR_];      // 22848 B : csum quarter-tile

  const int tid  = threadIdx.x;
  const int wg   = blockIdx.x;           // 256 = B * ND * 8 tiles
  const int b    = wg >> 5;
  const int rem  = wg & 31;
  const int di   = rem >> 3;             // dilation index
  const int tile = rem & 7;
  const int d    = 1 << di;
  const int pad  = 4 << di;              // (9-1)*d/2
  const int tstart = tile * TILE_;

  // Stage A[j, c*9+k] = comb[di][j][c] * w[j][k]   (rows 84..95 zero)
  for (int idx = tid; idx < MROWS_ * KDIM_; idx += 256) {
    int j = idx / KDIM_, col = idx % KDIM_;
    int c = col / 9,     k   = col % 9;
    float v = 0.f;
    if (j < NK_) v = comb[(di * NK_ + j) * C_ + c] * kers[j * 9 + k];
    sA[j * ASTR_ + col] = v;
  }
  // Stage x tile with halo (conv zero-padding realized here)
  for (int idx = tid; idx < C_ * XW_; idx += 256) {
    int c = idx / XW_, tt = idx % XW_;
    int t = tstart + tt - HALO_;
    sX[c * XW_ + tt] = (t >= 0 && t < L_) ? x[(b * C_ + c) * L_ + t] : 0.f;
  }
  __syncthreads();

  const int wave = tid >> 5;
  const int lane = tid & 31;
  const int lh   = lane & 15;
  const int hi   = lane >> 4;

  // Each thread owns up to 10 (j,f) PPV buckets, accumulated in registers.
  float partial[10];
  float pb[10];
  int   pj[10], puse[10];
#pragma unroll
  for (int s = 0; s < 10; ++s) {
    partial[s] = 0.f;
    int p = tid + s * 256;
    if (p < NK_ * NF_) {
      int j = p / NF_, f = p % NF_;
      pj[s]   = j;
      pb[s]   = biases[(di * NK_ + j) * NF_ + f];
      puse[s] = (((di + j) & 1) == 0);
    } else { pj[s] = -1; pb[s] = 0.f; puse[s] = 1; }
  }

  for (int q = 0; q < 4; ++q) {
    const int qbase = q * QT_;
    // ---- GEMM quarter: 6 m-tiles x 4 n-tiles = 24 tiles; 3 per wave ----
    for (int s = 0; s < 3; ++s) {
      int tt = wave + 8 * s;
      int m  = tt >> 2, n = tt & 3;
      v8f accv = {0.f, 0.f, 0.f, 0.f, 0.f, 0.f, 0.f, 0.f};
      int tloc = qbase + n * 16 + lh + HALO_;     // im2col column position in sX
#pragma unroll
      for (int kk = 0; kk < 18; ++kk) {
        // A fragment (16x4 f32): lanes 0-15 carry K0/K1, lanes 16-31 carry K2/K3
        int ka = kk * 4 + (hi ? 2 : 0);
        v2f av, bv;
        av.x = sA[(m * 16 + lh) * ASTR_ + ka];
        av.y = sA[(m * 16 + lh) * ASTR_ + ka + 1];
        // B fragment (4x16 f32): row r = c*9+k maps to x[t + (k-4)*d] of channel c
        int c0 = ka / 9,       k0 = ka % 9;
        int c1 = (ka + 1) / 9, k1 = (ka + 1) % 9;
        bv.x = sX[c0 * XW_ + tloc + (k0 - 4) * d];
        bv.y = sX[c1 * XW_ + tloc + (k1 - 4) * d];
        accv = __builtin_amdgcn_wmma_f32_16x16x4_f32(false, av, false, bv,
                                                     (short)0, accv, false, false);
      }
      // D layout: VGPR r -> (M = m*16 + r [+8 for hi half], N = n*16 + lh)
#pragma unroll
      for (int r = 0; r < 8; ++r) {
        int row = m * 16 + r + (hi ? 8 : 0);
        if (row < NK_) sC[row * CSTR_ + n * 16 + lh] = accv[r];
      }
    }
    __syncthreads();
    // ---- soft-PPV partial sums over this quarter ----
#pragma unroll
    for (int s = 0; s < 10; ++s) {
      int j = pj[s];
      if (j >= 0) {
        float bias = pb[s];
        int   uf   = puse[s];
        float sum  = 0.f;
        for (int tl = 0; tl < QT_; ++tl) {
          int   t  = tstart + qbase + tl;
          float v  = sC[j * CSTR_ + tl];
          float sg = 1.f / (1.f + __expf(bias - v));   // sigmoid(v - bias)
          float w  = (uf || (t >= pad && t < (L_ - pad))) ? 1.f : 0.f;
          sum += w * sg;
        }
        partial[s] += sum;
      }
    }
    __syncthreads();
  }

  // One global atomic per (b, di, j, f) bucket per workgroup (8-way contention only)
#pragma unroll
  for (int s = 0; s < 10; ++s) {
    int p = tid + s * 256;
    if (p < NK_ * NF_)
      atomicAdd(&acc[(b * ND_ + di) * NK_ * NF_ + p], partial[s]);
  }
}

// ---------------- normalize: divide by L (full) or L-2p (trimmed) ----------------
__global__ __launch_bounds__(256) void mr_norm(const float* __restrict__ acc,
                                               float* __restrict__ out, int n) {
  int idx = blockIdx.x * blockDim.x + threadIdx.x;
  if (idx >= n) return;
  int r  = idx % (ND_ * NK_ * NF_);
  int di = r / (NK_ * NF_);
  int j  = (r % (NK_ * NF_)) / NF_;
  int pad = 4 << di;
  int useFull = (((di + j) & 1) == 0);
  float cnt = useFull ? (float)L_ : (float)(L_ - 2 * pad);
  out[idx] = acc[idx] / cnt;
}

extern "C" void kernel_launch(void* const* d_in, const int* in_sizes, int n_in,
                              void* d_out, int out_size, void* d_ws, size_t ws_size,
                              hipStream_t stream) {
  const float* x      = (const float*)d_in[0];   // (8,8,2048)
  const float* kers   = (const float*)d_in[1];   // (8*84,1,9) tiled; first 84 used
  const float* comb   = (const float*)d_in[2];   // (4,84,8)
  const float* biases = (const float*)d_in[3];   // (4,84,30)
  float* acc = (float*)d_ws;                     // (8,4,84,30) f32 accumulator
  float* out = (float*)d_out;                    // (8, 4*84*30)

  const int total = B_ * ND_ * NK_ * NF_;        // 80640 == out_size
  mr_zero<<<(total + 255) / 256, 256, 0, stream>>>(acc, total);
  mr_main<<<B_ * ND_ * 8, 256, 0, stream>>>(x, kers, comb, biases, acc);
  mr_norm<<<(total + 255) / 256, 256, 0, stream>>>(acc, out, total);
}